// SGCNLayer_86723979640941
// MI455X (gfx1250) — compile-verified
//
#include <hip/hip_runtime.h>

#define C_IN   64
#define C_OUT  256
#define CK     1024
#define NB     2
#define NN     10000
#define NE     120000
#define EPB    16      // edges per block in the scatter kernel

typedef __attribute__((ext_vector_type(2))) float v2f;
typedef __attribute__((ext_vector_type(8))) float v8f;

// ---------------------------------------------------------------------------
// Kernel 1: initialize agg[b,n,k] = feats[b,n,k%64] * relu(b_dir[k])
// (self-contribution folded into the zero-init; also scrubs poisoned d_ws)
// ---------------------------------------------------------------------------
__global__ void sgcn_init_agg(const float* __restrict__ feats,
                              const float* __restrict__ b_dir,
                              float* __restrict__ agg) {
    size_t idx = (size_t)blockIdx.x * blockDim.x + threadIdx.x;
    size_t total = (size_t)NB * NN * CK;
    if (idx >= total) return;
    int    k    = (int)(idx & (CK - 1));
    size_t node = idx >> 10;                       // flattened b*N + n
    float  f    = feats[node * C_IN + (k & (C_IN - 1))];
    agg[idx]    = f * fmaxf(b_dir[k], 0.0f);
}

// ---------------------------------------------------------------------------
// Kernel 2: fused edge kernel.
//   aff_k = relu(d . W_dir[k] + b_dir[k]); msg = feats[src, k%64] * aff_k;
//   native-f32 atomic scatter-add into agg[dst]. W_dir/b_dir staged in LDS
//   (16 KB) and reused across EPB edges x both batches. agg (82 MB) is
//   L2-resident (192 MB L2), so atomic traffic never touches HBM.
// ---------------------------------------------------------------------------
__global__ void sgcn_edge_scatter(const float* __restrict__ feats,
                                  const float* __restrict__ pos,
                                  const int*   __restrict__ src,
                                  const int*   __restrict__ dst,
                                  const float* __restrict__ W_dir,
                                  const float* __restrict__ b_dir,
                                  float* __restrict__ agg) {
    __shared__ float sW[CK * 3];
    __shared__ float sb[CK];
    int t = threadIdx.x;
    for (int i = t; i < CK * 3; i += 256) sW[i] = W_dir[i];
    for (int i = t; i < CK;     i += 256) sb[i] = b_dir[i];
    __syncthreads();

    const int e0   = blockIdx.x * EPB;
    const int eend = (e0 + EPB < NE) ? (e0 + EPB) : NE;
    const int c    = t & (C_IN - 1);                // tile(): channel = k % 64

    for (int e = e0; e < eend; ++e) {
        const int s  = src[e];
        const int dn = dst[e];
#pragma unroll
        for (int b = 0; b < NB; ++b) {
            const float* posb  = pos   + (size_t)b * NN * 3;
            const float* featb = feats + (size_t)b * NN * C_IN;
            float*       arow  = agg   + ((size_t)b * NN + dn) * CK;

            const float dx = posb[s * 3 + 0] - posb[dn * 3 + 0];
            const float dy = posb[s * 3 + 1] - posb[dn * 3 + 1];
            const float dz = posb[s * 3 + 2] - posb[dn * 3 + 2];
            const float f  = featb[(size_t)s * C_IN + c];
#pragma unroll
            for (int j = 0; j < 4; ++j) {
                const int k = t + 256 * j;
                float a = fmaf(dx, sW[3 * k + 0],
                          fmaf(dy, sW[3 * k + 1],
                          fmaf(dz, sW[3 * k + 2], sb[k])));
                a = fmaxf(a, 0.0f) * f;
                // maps directly to global_atomic_add_f32 (no CAS loop)
                unsafeAtomicAdd(arow + k, a);
            }
        }
    }
}

// ---------------------------------------------------------------------------
// Kernel 3: out = relu(agg @ W_feat^T + b_feat) via V_WMMA_F32_16X16X4_F32.
// One wave = one 16x16 output tile; K looped in steps of 4 (256 WMMAs).
// A frag (16x4 f32, 2 VGPRs): lane m=lane&15, K-pair selected by lane>>4.
// B frag (4x16 f32, 2 VGPRs): B[k][o] = W_feat[o][k], same pairing.
// D (8 VGPRs): VGPR r, lane<16 -> M=r, N=lane; lane>=16 -> M=r+8, N=lane-16.
// No divergence: 20000 tiles == 2500 blocks * 8 waves exactly (EXEC all 1s).
// ---------------------------------------------------------------------------
__global__ void sgcn_gemm_wmma(const float* __restrict__ agg,
                               const float* __restrict__ W_feat,
                               const float* __restrict__ b_feat,
                               float* __restrict__ out) {
    const int lane = threadIdx.x & 31;
    const int warp = threadIdx.x >> 5;
    const int tile = blockIdx.x * 8 + warp;     // 0..19999
    const int tm   = tile >> 4;                 // 0..1249  (M tiles)
    const int tn   = tile & 15;                 // 0..15    (N tiles)
    const int row0 = tm << 4;
    const int col0 = tn << 4;
    const int half = lane >> 4;                 // K-pair / M-half selector
    const int lrow = lane & 15;

    const float* Aip = agg    + (size_t)(row0 + lrow) * CK + 2 * half;
    const float* Bip = W_feat + (size_t)(col0 + lrow) * CK + 2 * half;

    v8f acc = {};
#pragma unroll 8
    for (int k = 0; k < CK; k += 4) {
        v2f a  = *(const v2f*)(Aip + k);
        v2f bm = *(const v2f*)(Bip + k);
        // 8 args: (neg_a, A, neg_b, B, c_mod, C, reuse_a, reuse_b)
        acc = __builtin_amdgcn_wmma_f32_16x16x4_f32(
            false, a, false, bm, (short)0, acc, false, false);
    }

    const float bias = b_feat[col0 + lrow];
    const int   m0   = row0 + half * 8;
    const int   ncol = col0 + lrow;
#pragma unroll
    for (int r = 0; r < 8; ++r) {
        out[(size_t)(m0 + r) * C_OUT + ncol] = fmaxf(acc[r] + bias, 0.0f);
    }
}

// ---------------------------------------------------------------------------
extern "C" void kernel_launch(void* const* d_in, const int* in_sizes, int n_in,
                              void* d_out, int out_size, void* d_ws, size_t ws_size,
                              hipStream_t stream) {
    const float* feats  = (const float*)d_in[0];   // [B,N,64]
    const float* pos    = (const float*)d_in[1];   // [B,N,3]
    const int*   src    = (const int*)  d_in[2];   // [E]
    const int*   dst    = (const int*)  d_in[3];   // [E]
    const float* W_dir  = (const float*)d_in[4];   // [1024,3]
    const float* b_dir  = (const float*)d_in[5];   // [1024]
    const float* W_feat = (const float*)d_in[6];   // [256,1024]
    const float* b_feat = (const float*)d_in[7];   // [256]
    float*       out    = (float*)d_out;           // [B,N,256]
    float*       agg    = (float*)d_ws;            // [B,N,1024] = 81.92 MB

    (void)in_sizes; (void)n_in; (void)out_size; (void)ws_size;

    const size_t total = (size_t)NB * NN * CK;
    sgcn_init_agg<<<(unsigned)((total + 255) / 256), 256, 0, stream>>>(
        feats, b_dir, agg);

    sgcn_edge_scatter<<<(NE + EPB - 1) / EPB, 256, 0, stream>>>(
        feats, pos, src, dst, W_dir, b_dir, agg);

    const int tiles = (NB * NN / 16) * (C_OUT / 16);  // 20000
    sgcn_gemm_wmma<<<tiles / 8, 256, 0, stream>>>(
        agg, W_feat, b_feat, out);
}